// QuantumPatchEncoder_65481071398646
// MI455X (gfx1250) — compile-verified
//
#include <hip/hip_runtime.h>

typedef __attribute__((ext_vector_type(2))) float v2f;
typedef __attribute__((ext_vector_type(8))) float v8f;

#define NW 4
#define NL 2
#define BLOCK 256
#define SROW 20   // padded row (floats): 16B-aligned rows, conflict-free lane-half banking

// ---------------------------------------------------------------------------
// Kernel 1: build the batch-uniform circuit matrix W (16 x 32 fp32).
// Thread k propagates basis state |k> through the circuit -> column k of U.
// W[k][j]      = Re U[j,k]   (j = 0..15)
// W[k][16 + j] = Im U[j,k]
// ---------------------------------------------------------------------------
__global__ void qc_build_W(const float* __restrict__ params, float* __restrict__ W) {
    const int k = threadIdx.x;
    if (k >= 16) return;

    float sr[16], si[16];
#pragma unroll
    for (int j = 0; j < 16; ++j) { sr[j] = (j == k) ? 1.f : 0.f; si[j] = 0.f; }

#pragma unroll
    for (int layer = 0; layer < NL; ++layer) {
#pragma unroll
        for (int w = 0; w < NW; ++w) {
            // U = RX(theta_x) @ RZ(theta_z)
            const float tz = 0.5f * params[(layer * NW + w) * 2 + 0];
            const float tx = 0.5f * params[(layer * NW + w) * 2 + 1];
            const float ca = cosf(tz), sa = sinf(tz);
            const float c  = cosf(tx), s  = sinf(tx);
            const float u00r =  c * ca, u00i = -c * sa;
            const float u01r =  s * sa, u01i = -s * ca;
            const float u10r = -s * sa, u10i = -s * ca;
            const float u11r =  c * ca, u11i =  c * sa;
            const int bit = 8 >> w;   // wire w lives at bit (3-w)
#pragma unroll
            for (int j = 0; j < 16; ++j) {
                if (j & bit) continue;
                const int j1 = j | bit;
                const float v0r = sr[j],  v0i = si[j];
                const float v1r = sr[j1], v1i = si[j1];
                sr[j]  = u00r*v0r - u00i*v0i + u01r*v1r - u01i*v1i;
                si[j]  = u00r*v0i + u00i*v0r + u01r*v1i + u01i*v1r;
                sr[j1] = u10r*v0r - u10i*v0i + u11r*v1r - u11i*v1i;
                si[j1] = u10r*v0i + u10i*v0r + u11r*v1i + u11i*v1r;
            }
        }
        // CNOT ring: (0,1) (1,2) (2,3) (3,0): swap target bit where control==1
#pragma unroll
        for (int cw = 0; cw < NW; ++cw) {
            const int tw = (cw + 1) & 3;
            const int cb = 8 >> cw, tb = 8 >> tw;
#pragma unroll
            for (int j = 0; j < 16; ++j) {
                if ((j & cb) && !(j & tb)) {
                    const int j1 = j | tb;
                    float t0 = sr[j]; sr[j] = sr[j1]; sr[j1] = t0;
                    float t1 = si[j]; si[j] = si[j1]; si[j1] = t1;
                }
            }
        }
    }
#pragma unroll
    for (int j = 0; j < 16; ++j) {
        W[k * 32 + j]      = sr[j];
        W[k * 32 + 16 + j] = si[j];
    }
}

// ---------------------------------------------------------------------------
// Kernel 2: per thread build real init state s (16 floats, 4x b128 LDS store),
// per wave fp32 WMMA GEMM tiles, probs through LDS, per-thread signed
// reduction, coalesced float4 output.
// ---------------------------------------------------------------------------
__global__ void __launch_bounds__(BLOCK) qc_batch(const float* __restrict__ x,
                                                  const float* __restrict__ W,
                                                  float* __restrict__ out) {
    __shared__ float ldsS[BLOCK][SROW];   // [element][state index] (padded)
    __shared__ float ldsP[BLOCK][SROW];   // [element][prob index]  (padded)

    const int t    = threadIdx.x;
    const int lane = t & 31;
    const int wid  = t >> 5;           // wave id 0..7
    const int half = lane >> 4;        // 0 / 1  (K-split of A/B fragments)
    const int l16  = lane & 15;
    const int blockBase = blockIdx.x * BLOCK;

    // ---- per-element product state: s[j] = a0[b3] a1[b2] a2[b1] a3[b0] ----
    const float4 xv = ((const float4*)x)[blockBase + t];
    const float c0 = __cosf(0.5f * xv.x), s0 = __sinf(0.5f * xv.x);
    const float c1 = __cosf(0.5f * xv.y), s1 = __sinf(0.5f * xv.y);
    const float c2 = __cosf(0.5f * xv.z), s2 = __sinf(0.5f * xv.z);
    const float c3 = __cosf(0.5f * xv.w), s3 = __sinf(0.5f * xv.w);
    const float hi[4] = { c0*c1, c0*s1, s0*c1, s0*s1 };   // wires 0,1 -> bits 3,2
    const float lo[4] = { c2*c3, c2*s3, s2*c3, s2*s3 };   // wires 2,3 -> bits 1,0
    {
        float4* srow = (float4*)&ldsS[t][0];              // 80B rows -> 16B aligned
#pragma unroll
        for (int a = 0; a < 4; ++a)
            srow[a] = make_float4(hi[a]*lo[0], hi[a]*lo[1], hi[a]*lo[2], hi[a]*lo[3]);
    }

    // ---- resident B fragments (uniform W, 2 KB, L2-hot) ----
    // chunk kk covers K = 4kk..4kk+3 ; nt=0 -> Re cols, nt=1 -> Im cols
    v2f Bt[4][2];
#pragma unroll
    for (int kk = 0; kk < 4; ++kk) {
        const int r0 = kk * 4 + 2 * half;
#pragma unroll
        for (int nt = 0; nt < 2; ++nt) {
            Bt[kk][nt].x = W[r0       * 32 + nt * 16 + l16];
            Bt[kk][nt].y = W[(r0 + 1) * 32 + nt * 16 + l16];
        }
    }

    __syncthreads();

#pragma unroll
    for (int tile = 0; tile < 2; ++tile) {
        const int mbase = (wid * 2 + tile) * 16;   // 16 batch rows per tile

        // A fragments: lane l<16 -> rows M=l, K={4kk,4kk+1}; lanes 16-31 -> K={4kk+2,4kk+3}
        v2f A[4];
#pragma unroll
        for (int kk = 0; kk < 4; ++kk)
            A[kk] = *(const v2f*)&ldsS[mbase + l16][kk * 4 + 2 * half];   // ds_load_b64

        v8f accRe = {};
        v8f accIm = {};
#pragma unroll
        for (int kk = 0; kk < 4; ++kk) {
            accRe = __builtin_amdgcn_wmma_f32_16x16x4_f32(
                false, A[kk], false, Bt[kk][0], (short)0, accRe, false, false);
            accIm = __builtin_amdgcn_wmma_f32_16x16x4_f32(
                false, A[kk], false, Bt[kk][1], (short)0, accIm, false, false);
        }

        // ---- probs back to LDS: lane owns (elem = mbase + r + 8*half, j = l16) ----
        const int erow = mbase + 8 * half;
#pragma unroll
        for (int r = 0; r < 8; ++r) {
            const float re = accRe[r], im = accIm[r];
            ldsP[erow + r][l16] = re * re + im * im;   // b32, immediate offsets
        }
    }

    __syncthreads();

    // ---- per-thread signed reduction over own element's 16 probs ----
    const float4* prow = (const float4*)&ldsP[t][0];
    const float4 pa = prow[0], pb = prow[1], pc = prow[2], pd = prow[3];
    const float pj[16] = { pa.x, pa.y, pa.z, pa.w,  pb.x, pb.y, pb.z, pb.w,
                           pc.x, pc.y, pc.z, pc.w,  pd.x, pd.y, pd.z, pd.w };
    float q[8], rr[8];
#pragma unroll
    for (int i = 0; i < 8; ++i) { q[i] = pj[2*i] + pj[2*i+1]; rr[i] = pj[2*i] - pj[2*i+1]; }
    const float ev3 = ((rr[0]+rr[1]) + (rr[2]+rr[3])) + ((rr[4]+rr[5]) + (rr[6]+rr[7]));
    float u[4], v[4];
#pragma unroll
    for (int i = 0; i < 4; ++i) { u[i] = q[2*i] + q[2*i+1]; v[i] = q[2*i] - q[2*i+1]; }
    const float ev2 = (v[0] + v[1]) + (v[2] + v[3]);
    const float ev1 = (u[0] - u[1]) + (u[2] - u[3]);
    const float ev0 = (u[0] + u[1]) - (u[2] + u[3]);

    ((float4*)out)[blockBase + t] = make_float4(ev0, ev1, ev2, ev3);
}

// ---------------------------------------------------------------------------
extern "C" void kernel_launch(void* const* d_in, const int* in_sizes, int n_in,
                              void* d_out, int out_size, void* d_ws, size_t ws_size,
                              hipStream_t stream) {
    const float* x      = (const float*)d_in[0];   // (B, 4) fp32
    const float* params = (const float*)d_in[1];   // (2, 4, 2) fp32
    float* out = (float*)d_out;                    // (B, 4) fp32
    float* W   = (float*)d_ws;                     // 16 x 32 fp32 circuit matrix

    const int batch  = in_sizes[0] / 4;            // 1,048,576 (divisible by BLOCK)
    const int blocks = batch / BLOCK;

    qc_build_W<<<1, 16, 0, stream>>>(params, W);
    qc_batch<<<blocks, BLOCK, 0, stream>>>(x, W, out);
}